// VectorQuantizer_70729521431110
// MI455X (gfx1250) — compile-verified
//
#include <hip/hip_runtime.h>
#include <hip/hip_bf16.h>
#include <math.h>

typedef __attribute__((ext_vector_type(2))) float v2f;
typedef __attribute__((ext_vector_type(8))) float v8f;

#define K_DIM     131072
#define E_DIM     512
#define B_DIM     64
#define K_SPLIT   64
#define KC        (K_DIM / K_SPLIT)        // 2048 per wave
#define QN        ((size_t)B_DIM * K_DIM)  // 8388608

// ---------------- workspace layout (bytes) ----------------
// dot      : 64*512 f32   @ 0        (131072 B)
// norm     : 512 f32      @ 131072   (2048 B)
// counts   : 512 i32      @ 133120   (2048 B)
// idx      : 64 i32       @ 135168   (256 B)
// loss_acc : 1 f32        @ 135424

__global__ __launch_bounds__(256)
void vq_init(float* __restrict__ dot, float* __restrict__ norm,
             int* __restrict__ counts, float* __restrict__ loss_acc) {
    int i = blockIdx.x * 256 + threadIdx.x;
    if (i < B_DIM * E_DIM) dot[i] = 0.0f;
    if (i < E_DIM) { norm[i] = 0.0f; counts[i] = 0; }
    if (i == 0) *loss_acc = 0.0f;
}

// distances GEMM via V_WMMA_F32_16X16X4_F32.
// One wave owns ALL FOUR 16-row M-tiles (4 x v8f accumulators) so each B
// fragment of the 256MB codebook stream is loaded exactly once and feeds 4
// WMMAs (1.25 loads/WMMA). blockIdx.x = N-tile, blockIdx.y = K-group; the 4
// waves of a block take 4 consecutive K-splits and tree-reduce through LDS
// before the atomic scatter (cuts partial-sum atomics 4x vs per-wave scatter).
// ||e_n||^2 is folded into the same streaming pass (each W element is now
// visited by exactly one wave -> no divergent guard needed).
__global__ __launch_bounds__(128)
void vq_gemm(const float* __restrict__ X, const float* __restrict__ W,
             float* __restrict__ dot, float* __restrict__ norm) {
    __shared__ float red[4 * 4 * 8 * 32];   // [src_wave][m_tile][reg][lane] = 16KB

    const int lane = threadIdx.x;           // 0..31
    const int w    = threadIdx.y;           // 0..3 wave in block
    const int nt   = blockIdx.x;            // 0..31 N-tile
    const int kg   = blockIdx.y;            // 0..15 K-group
    const int k0   = (kg * 4 + w) * KC;

    // A frag (16x4 f32): lane L -> row m=L%16, k pair base 2*(L/16)
    // B frag (4x16 f32): lane L -> col n=L%16, same k pairing
    const int ncol  = (nt << 4) + (lane & 15);
    const int kpair = (lane >> 4) << 1;

    const float* __restrict__ Bp = W + (size_t)ncol * K_DIM + k0 + kpair;
    const float* __restrict__ Ap = X + (size_t)(lane & 15) * K_DIM + k0 + kpair;

    v8f c0 = {}, c1 = {}, c2 = {}, c3 = {};
    float nsum = 0.0f;

#pragma unroll 2
    for (int k = 0; k < KC; k += 4) {
        v2f b  = *(const v2f*)(Bp + k);
        v2f a0 = *(const v2f*)(Ap + k);
        v2f a1 = *(const v2f*)(Ap + (size_t)16 * K_DIM + k);
        v2f a2 = *(const v2f*)(Ap + (size_t)32 * K_DIM + k);
        v2f a3 = *(const v2f*)(Ap + (size_t)48 * K_DIM + k);
        nsum += b.x * b.x + b.y * b.y;
        c0 = __builtin_amdgcn_wmma_f32_16x16x4_f32(false, a0, false, b, (short)0, c0, false, false);
        c1 = __builtin_amdgcn_wmma_f32_16x16x4_f32(false, a1, false, b, (short)0, c1, false, false);
        c2 = __builtin_amdgcn_wmma_f32_16x16x4_f32(false, a2, false, b, (short)0, c2, false, false);
        c3 = __builtin_amdgcn_wmma_f32_16x16x4_f32(false, a3, false, b, (short)0, c3, false, false);
    }

    // stage partials: red[(w*4 + mt)*8 + r][lane]
#pragma unroll
    for (int r = 0; r < 8; ++r) {
        red[(((w * 4 + 0) * 8 + r) << 5) + lane] = c0[r];
        red[(((w * 4 + 1) * 8 + r) << 5) + lane] = c1[r];
        red[(((w * 4 + 2) * 8 + r) << 5) + lane] = c2[r];
        red[(((w * 4 + 3) * 8 + r) << 5) + lane] = c3[r];
    }
    __syncthreads();

    // wave w reduces m-tile w across the 4 source waves, then scatters.
    // C layout: reg r, lanes 0-15 -> M=r, lanes 16-31 -> M=r+8, N = lane&15.
    const int mbase = (w << 4) + ((lane >> 4) << 3);
#pragma unroll
    for (int r = 0; r < 8; ++r) {
        float s = 0.0f;
#pragma unroll
        for (int sw = 0; sw < 4; ++sw)
            s += red[(((sw * 4 + w) * 8 + r) << 5) + lane];
        atomicAdd(&dot[(size_t)(mbase + r) * E_DIM + ncol], s);
    }
    atomicAdd(&norm[ncol], nsum);
}

// per-row argmin of (||e||^2 - 2*dot), one-hot encodings, histogram counts
__global__ __launch_bounds__(512)
void vq_argmin(const float* __restrict__ dot, const float* __restrict__ norm,
               int* __restrict__ idx, int* __restrict__ counts,
               float* __restrict__ enc) {
    __shared__ float sval[E_DIM];
    __shared__ int   sidx[E_DIM];
    const int b = blockIdx.x;
    const int t = threadIdx.x;

    float d = norm[t] - 2.0f * dot[b * E_DIM + t];
    sval[t] = d; sidx[t] = t;
    __syncthreads();
    for (int s = 256; s > 0; s >>= 1) {
        if (t < s) {
            float v2 = sval[t + s]; int i2 = sidx[t + s];
            if (v2 < sval[t] || (v2 == sval[t] && i2 < sidx[t])) {
                sval[t] = v2; sidx[t] = i2;
            }
        }
        __syncthreads();
    }
    const int best = sidx[0];
    enc[b * E_DIM + t] = (t == best) ? 1.0f : 0.0f;
    if (t == 0) { idx[b] = best; atomicAdd(&counts[best], 1); }
}

// quantized = W[idx[b], :] (straight-through forward value) + MSE accumulation
__global__ __launch_bounds__(256)
void vq_quant(const float* __restrict__ X, const float* __restrict__ W,
              const int* __restrict__ idx, float* __restrict__ quant,
              float* __restrict__ loss_acc) {
    __shared__ float red[256];
    const size_t gid = (size_t)blockIdx.x * 256 + threadIdx.x;
    const int b = (int)(gid >> 17);        // / 131072
    const int k = (int)(gid & (K_DIM - 1));
    const float q = W[(size_t)idx[b] * K_DIM + k];
    const float x = X[gid];
    quant[gid] = q;
    const float df = q - x;
    red[threadIdx.x] = df * df;
    __syncthreads();
    for (int s = 128; s > 0; s >>= 1) {
        if (threadIdx.x < s) red[threadIdx.x] += red[threadIdx.x + s];
        __syncthreads();
    }
    if (threadIdx.x == 0) atomicAdd(loss_acc, red[0]);
}

// scalars: loss, perplexity, object_classes passthrough
__global__ __launch_bounds__(512)
void vq_scalars(const int* __restrict__ counts, const float* __restrict__ loss_acc,
                const int* __restrict__ obj, float* __restrict__ out_loss,
                float* __restrict__ out_perp, float* __restrict__ out_obj) {
    __shared__ float red[E_DIM];
    const int t = threadIdx.x;
    const float p = (float)counts[t] / (float)B_DIM;
    red[t] = -p * logf(p + 1e-10f);
    __syncthreads();
    for (int s = 256; s > 0; s >>= 1) {
        if (t < s) red[t] += red[t + s];
        __syncthreads();
    }
    if (t == 0) {
        *out_perp = expf(red[0]);
        *out_loss = (1.0f + 0.25f) * (*loss_acc) / (float)QN;
    }
    if (t < B_DIM) out_obj[t] = (float)obj[t];
}

extern "C" void kernel_launch(void* const* d_in, const int* in_sizes, int n_in,
                              void* d_out, int out_size, void* d_ws, size_t ws_size,
                              hipStream_t stream) {
    const float* X   = (const float*)d_in[0];   // [64,32,16,16,16] -> [64,131072]
    const int*   obj = (const int*)d_in[1];     // [64]
    const float* W   = (const float*)d_in[2];   // [512,131072]

    char* ws = (char*)d_ws;
    float* dot      = (float*)(ws + 0);
    float* norm     = (float*)(ws + 131072);
    int*   counts   = (int*)  (ws + 133120);
    int*   idx      = (int*)  (ws + 135168);
    float* loss_acc = (float*)(ws + 135424);

    float* out      = (float*)d_out;
    float* out_loss = out;                          // [0]
    float* out_q    = out + 1;                      // [1 .. 8388608]
    float* out_perp = out + 1 + QN;                 // [8388609]
    float* out_enc  = out + 2 + QN;                 // [8388610 ..]
    float* out_obj  = out + 2 + QN + B_DIM * E_DIM; // [8421378 ..]

    vq_init<<<dim3((B_DIM * E_DIM + 255) / 256), dim3(256), 0, stream>>>(
        dot, norm, counts, loss_acc);

    vq_gemm<<<dim3(E_DIM / 16, K_SPLIT / 4), dim3(32, 4), 0, stream>>>(
        X, W, dot, norm);

    vq_argmin<<<dim3(B_DIM), dim3(E_DIM), 0, stream>>>(
        dot, norm, idx, counts, out_enc);

    vq_quant<<<dim3((unsigned)(QN / 256)), dim3(256), 0, stream>>>(
        X, W, idx, out_q, loss_acc);

    vq_scalars<<<dim3(1), dim3(E_DIM), 0, stream>>>(
        counts, loss_acc, obj, out_loss, out_perp, out_obj);
}